// GCN_GAT_SpatialEncoder_9053791060568
// MI455X (gfx1250) — compile-verified
//
#include <hip/hip_runtime.h>
#include <hip/hip_bf16.h>

#define IN_CH 128
#define HID 64
#define HEADS 2
#define NEG_SLOPE 0.2f

typedef __attribute__((ext_vector_type(2))) float v2f;
typedef __attribute__((ext_vector_type(8))) float v8f;

typedef __attribute__((address_space(1))) int as1_int;
typedef __attribute__((address_space(3))) int as3_int;

__device__ __forceinline__ float leaky(float a) { return a >= 0.f ? a : NEG_SLOPE * a; }
// monotonic float<->uint mapping so atomicMax(u32) implements float max (incl. negatives)
__device__ __forceinline__ unsigned f2ord(float f) {
  unsigned u = __float_as_uint(f);
  return (u & 0x80000000u) ? ~u : (u | 0x80000000u);
}
__device__ __forceinline__ float ord2f(unsigned u) {
  return (u & 0x80000000u) ? __uint_as_float(u & 0x7fffffffu) : __uint_as_float(~u);
}

// ---------------------------------------------------------------------------
// WMMA fp32 GEMM: C[M,NCOLS] = A[M,K] @ B[K,NCOLS], all row-major fp32.
// - B is staged once per block into LDS via async global->LDS copies
//   (GLOBAL_LOAD_ASYNC_TO_LDS_B32, per-lane scatter), pre-swizzled into WMMA
//   fragment-pair order so each hot-loop B read is a single ds_load_b64.
// - Each wave computes NSTRIP strips of 16 rows x NCOLS (NCOLS/16 WMMA tiles),
//   accumulating with V_WMMA_F32_16X16X4_F32 over K in steps of 4.
// Fragment layouts per CDNA5 ISA 7.12.2:
//   A 16x4:  lane<16 -> M=lane, K {0,1};  lane>=16 -> M=lane-16, K {2,3}
//   B 4x16:  elem j, half h -> K = 2h+j, N = lane%16
//   C 16x16: VGPR j -> M=j (lanes 0-15) / M=j+8 (lanes 16-31), N = lane%16
// ---------------------------------------------------------------------------
template <int K, int NCOLS, int NSTRIP>
__global__ void wmma_gemm_kernel(const float* __restrict__ A,
                                 const float* __restrict__ Bg,
                                 float* __restrict__ C, int M) {
  __shared__ float Bs[K * NCOLS];  // swizzled: ((kk*2+h)*NCOLS + col)*2 + j

  // ---- stage B into LDS (all waves participate, then barrier) ----
  constexpr int TOT = K * NCOLS;
  for (int i = (int)threadIdx.x; i < TOT; i += (int)blockDim.x) {
    const int row = i / NCOLS, col = i % NCOLS;
    const int idx = (((row >> 2) * 2 + ((row >> 1) & 1)) * NCOLS + col) * 2 + (row & 1);
#if defined(__gfx1250__) && __has_builtin(__builtin_amdgcn_global_load_async_to_lds_b32)
    __builtin_amdgcn_global_load_async_to_lds_b32((as1_int*)(Bg + i), (as3_int*)(&Bs[idx]),
                                                  0, 0);
#else
    Bs[idx] = Bg[i];
#endif
  }
#if defined(__gfx1250__) && __has_builtin(__builtin_amdgcn_global_load_async_to_lds_b32)
#if __has_builtin(__builtin_amdgcn_s_wait_asynccnt)
  __builtin_amdgcn_s_wait_asynccnt(0);
#else
  asm volatile("s_wait_asynccnt 0x0" ::: "memory");
#endif
#endif
  __syncthreads();

  const int lane = (int)(threadIdx.x & 31u);
  const int half = lane >> 4;
  const int r = lane & 15;
  const int waveG = (int)(blockIdx.x * (blockDim.x >> 5) + (threadIdx.x >> 5));
  constexpr int NT = NCOLS / 16;

  for (int s = 0; s < NSTRIP; ++s) {
    const int rowBase = (waveG * NSTRIP + s) * 16;
    if (rowBase >= M) return;  // wave-uniform: EXEC stays all-ones around WMMA

    v8f acc[NT];
#pragma unroll
    for (int t = 0; t < NT; ++t) acc[t] = (v8f){0.f, 0.f, 0.f, 0.f, 0.f, 0.f, 0.f, 0.f};

    int arow = rowBase + r;
    if (arow >= M) arow = M - 1;  // clamp (loads only; stores guarded)
    const float* ap = A + (size_t)arow * K;

    for (int kk = 0; kk < K / 4; ++kk) {
      const v2f a = *(const v2f*)(ap + kk * 4 + 2 * half);
#pragma unroll
      for (int t = 0; t < NT; ++t) {
        const int col = t * 16 + r;
        const v2f b = *(const v2f*)(&Bs[((kk * 2 + half) * NCOLS + col) * 2]);
        acc[t] = __builtin_amdgcn_wmma_f32_16x16x4_f32(false, a, false, b, (short)0,
                                                       acc[t], false, false);
      }
    }

    if (rowBase + 16 <= M) {  // fast path: full tile, no per-row guards
#pragma unroll
      for (int t = 0; t < NT; ++t) {
        float* cp = C + (size_t)(rowBase + 8 * half) * NCOLS + t * 16 + r;
#pragma unroll
        for (int j = 0; j < 8; ++j) cp[(size_t)j * NCOLS] = acc[t][j];
      }
    } else {
#pragma unroll
      for (int t = 0; t < NT; ++t) {
        const int col = t * 16 + r;
#pragma unroll
        for (int j = 0; j < 8; ++j) {
          const int row = rowBase + j + 8 * half;
          if (row < M) C[(size_t)row * NCOLS + col] = acc[t][j];
        }
      }
    }
  }
}

// ------------------------- GCN normalization ------------------------------
__global__ void k_deg_init(float* deg, int n) {
  int t = blockIdx.x * blockDim.x + threadIdx.x;
  if (t < n) deg[t] = 1.0f;  // self-loop weight
}
__global__ void k_deg_acc(const int* __restrict__ dst, const float* __restrict__ ew,
                          float* deg, int e) {
  int t = blockIdx.x * blockDim.x + threadIdx.x;
  if (t < e) atomicAdd(&deg[dst[t]], ew[t]);
}
__global__ void k_deg_inv(float* deg, int n) {
  int t = blockIdx.x * blockDim.x + threadIdx.x;
  if (t < n) {
    float x = deg[t];
    deg[t] = x > 0.f ? rsqrtf(fmaxf(x, 1e-12f)) : 0.f;
  }
}
// per-edge symmetric norm: enorm[e] = dinv[src]*ew*dinv[dst]
__global__ void k_edge_norm(const int* __restrict__ src, const int* __restrict__ dst,
                            const float* __restrict__ ew, const float* __restrict__ dinv,
                            float* enorm, int e) {
  int t = blockIdx.x * blockDim.x + threadIdx.x;
  if (t < e) enorm[t] = dinv[src[t]] * ew[t] * dinv[dst[t]];
}
// self-loop contribution: hg[n][c] = dinv[n]^2 * h1[n][c]
__global__ void k_gcn_self(const float* __restrict__ h1, const float* __restrict__ dinv,
                           float* __restrict__ hg, int n) {
  int t = blockIdx.x * blockDim.x + threadIdx.x;
  if (t >= n * 16) return;
  int node = t >> 4, c4 = (t & 15) << 2;
  float w = dinv[node];
  w *= w;
  const float4 v = *(const float4*)(h1 + (size_t)node * HID + c4);
  *(float4*)(hg + (size_t)node * HID + c4) = make_float4(w * v.x, w * v.y, w * v.z, w * v.w);
}
__global__ void k_gcn_edge(const int* __restrict__ src, const int* __restrict__ dst,
                           const float* __restrict__ enorm, const float* __restrict__ h1,
                           float* hg, int e) {
  int t = blockIdx.x * blockDim.x + threadIdx.x;
  if (t >= e * 16) return;
  int ed = t >> 4, c4 = (t & 15) << 2;
  int s = src[ed], d = dst[ed];
  float nrm = enorm[ed];
  const float4 v = *(const float4*)(h1 + (size_t)s * HID + c4);
  float* op = hg + (size_t)d * HID + c4;
  atomicAdd(op + 0, nrm * v.x);
  atomicAdd(op + 1, nrm * v.y);
  atomicAdd(op + 2, nrm * v.z);
  atomicAdd(op + 3, nrm * v.w);
}
__global__ void k_bias_relu(float* hg, const float* __restrict__ b, int n) {
  int t = blockIdx.x * blockDim.x + threadIdx.x;
  if (t >= n * HID) return;
  hg[t] = fmaxf(hg[t] + b[t & (HID - 1)], 0.f);
}

// ------------------------------ GAT ---------------------------------------
// per (node, head): attention scores + init segment-max with self-loop alpha
__global__ void k_scores(const float* __restrict__ g, const float* __restrict__ a_src,
                         const float* __restrict__ a_dst, float* ss, float* sd,
                         unsigned* amaxo, int n) {
  int t = blockIdx.x * blockDim.x + threadIdx.x;
  if (t >= n * 2) return;
  int node = t >> 1, hd = t & 1;
  const float* gp = g + (size_t)node * (HEADS * HID) + hd * HID;
  const float* a1 = a_src + hd * HID;
  const float* a2 = a_dst + hd * HID;
  float s = 0.f, d = 0.f;
  for (int c = 0; c < HID; ++c) {
    float v = gp[c];
    s += v * a1[c];
    d += v * a2[c];
  }
  ss[t] = s;
  sd[t] = d;
  amaxo[t] = f2ord(leaky(s + d));  // self-loop seeds the segment max
}
__global__ void k_edge_max(const int* __restrict__ src, const int* __restrict__ dst,
                           const float* __restrict__ ss, const float* __restrict__ sd,
                           unsigned* amaxo, int e) {
  int t = blockIdx.x * blockDim.x + threadIdx.x;
  if (t >= e * 2) return;
  int ed = t >> 1, hd = t & 1;
  int s = src[ed], d = dst[ed];
  float a = leaky(ss[s * 2 + hd] + sd[d * 2 + hd]);
  atomicMax(&amaxo[d * 2 + hd], f2ord(a));
}
__global__ void k_denom_init(const float* __restrict__ ss, const float* __restrict__ sd,
                             const unsigned* __restrict__ amaxo, float* denom, int n) {
  int t = blockIdx.x * blockDim.x + threadIdx.x;
  if (t >= n * 2) return;
  float a = leaky(ss[t] + sd[t]);
  denom[t] = expf(a - ord2f(amaxo[t]));  // self-loop term
}
__global__ void k_edge_denom(const int* __restrict__ src, const int* __restrict__ dst,
                             const float* __restrict__ ss, const float* __restrict__ sd,
                             const unsigned* __restrict__ amaxo, float* denom, int e) {
  int t = blockIdx.x * blockDim.x + threadIdx.x;
  if (t >= e * 2) return;
  int ed = t >> 1, hd = t & 1;
  int s = src[ed], d = dst[ed];
  int nh = d * 2 + hd;
  float a = leaky(ss[s * 2 + hd] + sd[nh]);
  atomicAdd(&denom[nh], expf(a - ord2f(amaxo[nh])));
}
// per-(edge,head) softmax coefficient, materialized once
__global__ void k_edge_coef(const int* __restrict__ src, const int* __restrict__ dst,
                            const float* __restrict__ ss, const float* __restrict__ sd,
                            const unsigned* __restrict__ amaxo,
                            const float* __restrict__ denom, float* ecoef, int e) {
  int t = blockIdx.x * blockDim.x + threadIdx.x;
  if (t >= e * 2) return;
  int ed = t >> 1, hd = t & 1;
  int nh = dst[ed] * 2 + hd;
  float a = leaky(ss[src[ed] * 2 + hd] + sd[nh]);
  ecoef[t] = expf(a - ord2f(amaxo[nh])) / fmaxf(denom[nh], 1e-16f);
}
__global__ void k_agg_init(const float* __restrict__ g, const float* __restrict__ ss,
                           const float* __restrict__ sd, const unsigned* __restrict__ amaxo,
                           const float* __restrict__ denom, float* outacc, int n) {
  int t = blockIdx.x * blockDim.x + threadIdx.x;
  if (t >= n * 32) return;
  int node = t >> 5, hd = (t >> 4) & 1, q = t & 15;
  int nh = node * 2 + hd;
  float a = leaky(ss[nh] + sd[nh]);
  float coef = expf(a - ord2f(amaxo[nh])) / fmaxf(denom[nh], 1e-16f);
  size_t off = (size_t)node * (HEADS * HID) + hd * HID + q * 4;
  const float4 v = *(const float4*)(g + off);
  *(float4*)(outacc + off) = make_float4(coef * v.x, coef * v.y, coef * v.z, coef * v.w);
}
__global__ void k_edge_agg(const int* __restrict__ src, const int* __restrict__ dst,
                           const float* __restrict__ g, const float* __restrict__ ecoef,
                           float* outacc, int e) {
  int t = blockIdx.x * blockDim.x + threadIdx.x;
  if (t >= e * 32) return;
  int ed = t >> 5, hd = (t >> 4) & 1, q = t & 15;
  int s = src[ed], d = dst[ed];
  float coef = ecoef[ed * 2 + hd];
  const float4 v = *(const float4*)(g + (size_t)s * (HEADS * HID) + hd * HID + q * 4);
  float* op = outacc + (size_t)d * (HEADS * HID) + hd * HID + q * 4;
  atomicAdd(op + 0, coef * v.x);
  atomicAdd(op + 1, coef * v.y);
  atomicAdd(op + 2, coef * v.z);
  atomicAdd(op + 3, coef * v.w);
}
__global__ void k_final(const float* __restrict__ outacc, const float* __restrict__ gat_b,
                        float* out, int n) {
  int t = blockIdx.x * blockDim.x + threadIdx.x;
  if (t >= n * HID) return;
  int node = t >> 6, c = t & 63;
  float v = 0.5f * (outacc[(size_t)node * 128 + c] + outacc[(size_t)node * 128 + 64 + c]) +
            gat_b[c];
  out[t] = fmaxf(v, 0.f);
}

extern "C" void kernel_launch(void* const* d_in, const int* in_sizes, int n_in,
                              void* d_out, int out_size, void* d_ws, size_t ws_size,
                              hipStream_t stream) {
  const float* x = (const float*)d_in[0];
  const int* eidx = (const int*)d_in[1];
  const float* ew = (const float*)d_in[2];
  const float* gcn_w = (const float*)d_in[3];
  const float* gcn_b = (const float*)d_in[4];
  const float* gat_w = (const float*)d_in[5];
  const float* att_s = (const float*)d_in[6];
  const float* att_d = (const float*)d_in[7];
  const float* gat_b = (const float*)d_in[8];
  float* out = (float*)d_out;

  const int n = in_sizes[0] / IN_CH;
  const int e = in_sizes[1] / 2;
  const int* src = eidx;
  const int* dst = eidx + e;

  // workspace (floats): dinv[n] | h1[n*64] | hg[n*64] | g[n*128] |
  //                     ss[2n] | sd[2n] | amaxo[2n] | denom[2n] | ecoef[2e]
  // outacc reuses the contiguous h1|hg region (n*128) once g is computed;
  // enorm (e floats, dead after GCN scatter) aliases ecoef.
  float* ws = (float*)d_ws;
  float* dinv = ws;
  float* h1 = dinv + n;
  float* hg = h1 + (size_t)n * HID;
  float* g = hg + (size_t)n * HID;
  float* ss = g + (size_t)n * HEADS * HID;
  float* sd = ss + (size_t)n * 2;
  unsigned* amaxo = (unsigned*)(sd + (size_t)n * 2);
  float* denom = (float*)(amaxo + (size_t)n * 2);
  float* ecoef = denom + (size_t)n * 2;
  float* enorm = ecoef;  // alias: non-overlapping lifetimes
  float* outacc = h1;

  const int B = 256;
  auto nb = [&](long long total) { return (int)((total + B - 1) / B); };

  // ---- GCN degree norm ----
  k_deg_init<<<nb(n), B, 0, stream>>>(dinv, n);
  k_deg_acc<<<nb(e), B, 0, stream>>>(dst, ew, dinv, e);
  k_deg_inv<<<nb(n), B, 0, stream>>>(dinv, n);
  k_edge_norm<<<nb(e), B, 0, stream>>>(src, dst, ew, dinv, enorm, e);

  // ---- GCN GEMM: h1 = x @ gcn_w (WMMA fp32, B staged via async->LDS) ----
  constexpr int NSTRIP = 4;
  const long long strips = (n + 15) / 16;
  const long long wavesNeeded = (strips + NSTRIP - 1) / NSTRIP;
  const int gemmBlocks = (int)((wavesNeeded + 7) / 8);  // 8 waves per 256-thread block
  wmma_gemm_kernel<IN_CH, HID, NSTRIP><<<gemmBlocks, B, 0, stream>>>(x, gcn_w, h1, n);

  // ---- GCN aggregate + bias + relu (into hg) ----
  k_gcn_self<<<nb((long long)n * 16), B, 0, stream>>>(h1, dinv, hg, n);
  k_gcn_edge<<<nb((long long)e * 16), B, 0, stream>>>(src, dst, enorm, h1, hg, e);
  k_bias_relu<<<nb((long long)n * HID), B, 0, stream>>>(hg, gcn_b, n);

  // ---- GAT GEMM: g = hg @ gat_w (WMMA fp32) ----
  wmma_gemm_kernel<HID, HEADS * HID, NSTRIP><<<gemmBlocks, B, 0, stream>>>(hg, gat_w, g, n);

  // ---- GAT attention ----
  k_scores<<<nb((long long)n * 2), B, 0, stream>>>(g, att_s, att_d, ss, sd, amaxo, n);
  k_edge_max<<<nb((long long)e * 2), B, 0, stream>>>(src, dst, ss, sd, amaxo, e);
  k_denom_init<<<nb((long long)n * 2), B, 0, stream>>>(ss, sd, amaxo, denom, n);
  k_edge_denom<<<nb((long long)e * 2), B, 0, stream>>>(src, dst, ss, sd, amaxo, denom, e);
  k_edge_coef<<<nb((long long)e * 2), B, 0, stream>>>(src, dst, ss, sd, amaxo, denom, ecoef, e);

  // ---- GAT aggregate (outacc reuses h1|hg region; hg fully consumed above) ----
  k_agg_init<<<nb((long long)n * 32), B, 0, stream>>>(g, ss, sd, amaxo, denom, outacc, n);
  k_edge_agg<<<nb((long long)e * 32), B, 0, stream>>>(src, dst, g, ecoef, outacc, e);

  // ---- head mean + bias + relu -> d_out ----
  k_final<<<nb((long long)n * HID), B, 0, stream>>>(outacc, gat_b, out, n);
}